// FullyConnected_19473381720424
// MI455X (gfx1250) — compile-verified
//
#include <hip/hip_runtime.h>
#include <stdint.h>

// MAM fully-connected: out[m,n] = max_k(x[m,k]*w[n,k]) + min_k(x[m,k]*w[n,k]) + bias[n]
// x: [8192,1024] f32, weight: [1024,1024] f32 (row-major [N,K]), bias: [1024] f32.
// VALU-bound (~25.8G f32 mul/max/min ops vs ~70MB traffic); LDS-tiled with TDM
// async tile loads (tensor_load_to_lds + s_wait_tensorcnt) and 4x4 register tiles.

#define M_DIM 8192
#define K_DIM 1024
#define N_DIM 1024
#define BM 64
#define BN 64
#define BK 32
#define LDS_STRIDE 36          // 32 dwords payload + 4 dwords TDM pad per row
#define NT (K_DIM / BK)        // 32 K-tiles

#if defined(__has_builtin)
#if __has_builtin(__builtin_amdgcn_tensor_load_to_lds) && \
    __has_builtin(__builtin_amdgcn_s_wait_tensorcnt)
#define USE_TDM 1
#endif
#endif
#ifndef USE_TDM
#define USE_TDM 0
#endif

typedef __attribute__((ext_vector_type(4))) unsigned int u32x4;
typedef __attribute__((ext_vector_type(4))) int i32x4;
typedef __attribute__((ext_vector_type(8))) int i32x8;

#if USE_TDM
// Issue one 2D TDM tile load: tile = 64 rows x 32 dwords, row stride K_DIM elems,
// data_size = 4B, LDS padding 4 dwords every 32 dwords (-> 36-dword LDS rows).
__device__ __forceinline__ void tdm_load_tile(uint32_t lds_byte_addr,
                                              const float* gptr,
                                              uint32_t tensor_d1) {
  const uint64_t ga = (uint64_t)(uintptr_t)gptr;

  u32x4 g0;
  g0.x = 1u;                                   // count=1, is_restore=0, no gather
  g0.y = lds_byte_addr;                        // lds_addr (bytes)
  g0.z = (uint32_t)ga;                         // global_addr[31:0]
  g0.w = (uint32_t)((ga >> 32) & 0x01FFFFFFu)  // global_addr[56:32]
         | (2u << 30);                         // type = 2 ("image")

  i32x8 g1;
  g1[0] = (int)((2u << 16)     // data_size: 4 bytes
              | (1u << 20)     // pad_enable
              | (4u << 22)     // pad_interval code 4 -> every 32 dwords
              | (3u << 25));   // pad_amount  code 3 -> 4 dwords
  g1[1] = (int)(((uint32_t)K_DIM & 0xFFFFu) << 16);                 // tensor_dim0 lo16
  g1[2] = (int)((((uint32_t)K_DIM >> 16) & 0xFFFFu)                 // tensor_dim0 hi16
              | ((tensor_d1 & 0xFFFFu) << 16));                     // tensor_dim1 lo16
  g1[3] = (int)(((tensor_d1 >> 16) & 0xFFFFu)                       // tensor_dim1 hi16
              | ((uint32_t)BK << 16));                              // tile_dim0 = 32
  g1[4] = (int)(uint32_t)BM;                                        // tile_dim1 = 64, tile_dim2 = 0
  g1[5] = (int)K_DIM;                                               // tensor_dim0_stride lo32
  g1[6] = 0;                                                        // stride0 hi16, stride1 lo16
  g1[7] = 0;

  const i32x4 gz = {0, 0, 0, 0};
#if defined(__clang_major__) && __clang_major__ >= 23
  const i32x8 gz8 = {0, 0, 0, 0, 0, 0, 0, 0};
  __builtin_amdgcn_tensor_load_to_lds(g0, g1, gz, gz, gz8, 0);
#else
  __builtin_amdgcn_tensor_load_to_lds(g0, g1, gz, gz, 0);
#endif
}
#endif

__device__ __forceinline__ void mam_tile(const float* __restrict__ Xb,
                                         const float* __restrict__ Wb,
                                         int ty, int tx,
                                         float mx[16], float mn[16]) {
#pragma unroll
  for (int k = 0; k < BK; k += 4) {
    float4 xv[4], wv[4];
#pragma unroll
    for (int i = 0; i < 4; ++i)
      xv[i] = *(const float4*)(Xb + (ty * 4 + i) * LDS_STRIDE + k);
#pragma unroll
    for (int j = 0; j < 4; ++j)
      wv[j] = *(const float4*)(Wb + (tx * 4 + j) * LDS_STRIDE + k);
#pragma unroll
    for (int i = 0; i < 4; ++i) {
#pragma unroll
      for (int j = 0; j < 4; ++j) {
        const int a = i * 4 + j;
        const float p0 = xv[i].x * wv[j].x;
        const float p1 = xv[i].y * wv[j].y;
        const float p2 = xv[i].z * wv[j].z;
        const float p3 = xv[i].w * wv[j].w;
        // written to encourage v_max3_num_f32 / v_min3_num_f32 fusion
        mx[a] = fmaxf(fmaxf(mx[a], p0), p1);
        mx[a] = fmaxf(fmaxf(mx[a], p2), p3);
        mn[a] = fminf(fminf(mn[a], p0), p1);
        mn[a] = fminf(fminf(mn[a], p2), p3);
      }
    }
  }
}

__global__ __launch_bounds__(256)
void mam_fc_kernel(const float* __restrict__ x,
                   const float* __restrict__ w,
                   const float* __restrict__ bias,
                   float* __restrict__ out) {
  __shared__ __align__(16) float Xl[2][BM * LDS_STRIDE];
  __shared__ __align__(16) float Wl[2][BN * LDS_STRIDE];

  const int tid = (int)threadIdx.x;
  const int tx = tid & 15;   // 4-col group in N
  const int ty = tid >> 4;   // 4-row group in M
  const int m0 = (int)blockIdx.y * BM;
  const int n0 = (int)blockIdx.x * BN;

  float mx[16], mn[16];
#pragma unroll
  for (int i = 0; i < 16; ++i) {
    mx[i] = -__builtin_inff();
    mn[i] = __builtin_inff();
  }

#if USE_TDM
  const bool issuer = (tid < 32);  // wave 0 issues all TDM descriptors
  if (issuer) {
    tdm_load_tile((uint32_t)(uintptr_t)(void*)Xl[0], x + (size_t)m0 * K_DIM, M_DIM);
    tdm_load_tile((uint32_t)(uintptr_t)(void*)Wl[0], w + (size_t)n0 * K_DIM, N_DIM);
    __builtin_amdgcn_s_wait_tensorcnt(0);
  }
  __syncthreads();

  for (int t = 0; t < NT; ++t) {
    const int cur = t & 1;
    if (issuer && (t + 1) < NT) {
      const int k1 = (t + 1) * BK;
      tdm_load_tile((uint32_t)(uintptr_t)(void*)Xl[cur ^ 1],
                    x + (size_t)m0 * K_DIM + k1, M_DIM);
      tdm_load_tile((uint32_t)(uintptr_t)(void*)Wl[cur ^ 1],
                    w + (size_t)n0 * K_DIM + k1, N_DIM);
    }
    mam_tile(Xl[cur], Wl[cur], ty, tx, mx, mn);
    __syncthreads();                                 // all waves done reading `cur`
    if (issuer) __builtin_amdgcn_s_wait_tensorcnt(0); // next buffer landed
    __syncthreads();                                 // publish next buffer
  }
#else
  // Fallback path (no TDM builtin on this toolchain): cooperative sync loads.
  for (int t = 0; t < NT; ++t) {
    const int cur = t & 1;
    const int k0 = t * BK;
    for (int idx = tid; idx < BM * BK; idx += 256) {
      const int r = idx >> 5;        // BK == 32
      const int c = idx & (BK - 1);
      Xl[cur][r * LDS_STRIDE + c] = x[(size_t)(m0 + r) * K_DIM + k0 + c];
      Wl[cur][r * LDS_STRIDE + c] = w[(size_t)(n0 + r) * K_DIM + k0 + c];
    }
    __syncthreads();
    mam_tile(Xl[cur], Wl[cur], ty, tx, mx, mn);
    __syncthreads();
  }
#endif

  const float4 bv = *(const float4*)(bias + n0 + tx * 4);
#pragma unroll
  for (int i = 0; i < 4; ++i) {
    const int m = m0 + ty * 4 + i;
    float4 o;
    o.x = mx[i * 4 + 0] + mn[i * 4 + 0] + bv.x;
    o.y = mx[i * 4 + 1] + mn[i * 4 + 1] + bv.y;
    o.z = mx[i * 4 + 2] + mn[i * 4 + 2] + bv.z;
    o.w = mx[i * 4 + 3] + mn[i * 4 + 3] + bv.w;
    *(float4*)(out + (size_t)m * N_DIM + n0 + tx * 4) = o;
  }
}

extern "C" void kernel_launch(void* const* d_in, const int* in_sizes, int n_in,
                              void* d_out, int out_size, void* d_ws, size_t ws_size,
                              hipStream_t stream) {
  (void)in_sizes; (void)n_in; (void)d_ws; (void)ws_size; (void)out_size;
  const float* x    = (const float*)d_in[0];
  const float* wgt  = (const float*)d_in[1];
  const float* bias = (const float*)d_in[2];
  float* out = (float*)d_out;

  dim3 grid(N_DIM / BN, M_DIM / BM);   // (16, 128) blocks
  dim3 block(256);                     // 8 wave32s
  mam_fc_kernel<<<grid, block, 0, stream>>>(x, wgt, bias, out);
}